// KPLayer_67027259621816
// MI455X (gfx1250) — compile-verified
//
#include <hip/hip_runtime.h>
#include <hip/hip_bf16.h>

#define B_DIM  256
#define N_DIM  512
#define E_DIM  128
#define T_DIM  511      // N-1 (valid back/fore pairs)
#define M_FORE 96
#define S_TOT  607      // T_DIM + M_FORE : number of generated states s_1..s_607

#define CH_ROWS 17      // staged rows per K-chunk (16 k-steps + 1 fore row)
#define NCHUNK  32      // 32 chunks x 16 = 512 k slots (k=511 zero-padded)

typedef float v2f __attribute__((ext_vector_type(2)));
typedef float v8f __attribute__((ext_vector_type(8)));
typedef int   v4i __attribute__((vector_size(16)));   // matches builtin param
union V8 { v8f v; float f[8]; };

static __device__ __forceinline__ v8f wmma_f32x4(v2f a, v2f b, v8f c) {
  // D(16x16,f32) = A(16x4,f32) x B(4x16,f32) + C
  return __builtin_amdgcn_wmma_f32_16x16x4_f32(false, a, false, b, (short)0, c, false, false);
}

#if __has_builtin(__builtin_amdgcn_global_load_async_to_lds_b128)
#define HAVE_ASYNC_LDS 1
#else
#define HAVE_ASYNC_LDS 0
#endif

static __device__ __forceinline__ void wait_async0() {
#if __has_builtin(__builtin_amdgcn_s_wait_asynccnt)
  __builtin_amdgcn_s_wait_asynccnt(0);
#else
  asm volatile("s_wait_asynccnt 0" ::: "memory");
#endif
}

// Cooperative stage of nrows contiguous z-rows (128 floats each) into LDS.
// Async DMA path (ASYNCcnt, GLOBAL_LOAD_ASYNC_TO_LDS_B128) when available;
// vectorized sync copy otherwise. Generic pointers: clang inserts the
// implicit addrspace casts to the builtin's __device__/__shared__ params.
static __device__ __forceinline__ void stage_rows(float* dst,
                                                  const float* src,
                                                  int nrows, int tid) {
  const int nvec = nrows * (E_DIM / 4);
#if HAVE_ASYNC_LDS
  for (int i = tid; i < nvec; i += 256)
    __builtin_amdgcn_global_load_async_to_lds_b128(
        (v4i*)(src + i * 4), (v4i*)(dst + i * 4), 0, 0);
#else
  for (int i = tid; i < nvec; i += 256)
    *(float4*)(dst + i * 4) = *(const float4*)(src + i * 4);
#endif
}

// ---------------------------------------------------------------------------
// Kernel 0: zero the global NaN flag
// ---------------------------------------------------------------------------
__global__ void zero_flag_kernel(int* flag) { *flag = 0; }

// ---------------------------------------------------------------------------
// Kernel 1: per-batch G = Zb^T Zb , C = Zb^T Zf  via V_WMMA_F32_16X16X4_F32.
// grid = (B, E/16), block = 256 (8 waves); wave w -> output col-tile w,
// blockIdx.y -> output row tile. Z rows are double-buffer staged into LDS
// with async B128 copies issued one chunk ahead so the DMA overlaps WMMA.
// ---------------------------------------------------------------------------
__global__ __launch_bounds__(256)
void gram_kernel(const float* __restrict__ z,
                 float* __restrict__ Gout, float* __restrict__ Cout) {
  __shared__ __align__(16) float sbuf[2][CH_ROWS * E_DIM];

  const int b    = blockIdx.x;
  const int ti   = blockIdx.y;          // output row tile (0..7)
  const int tid  = threadIdx.x;
  const int wave = tid >> 5;            // output col tile (0..7)
  const int lane = tid & 31;
  const int m    = lane & 15;
  const int kb   = (lane >> 4) << 1;    // 0 or 2 (K sub-offset per A/B layout)
  const int ca   = ti * 16 + m;         // A column inside a z row
  const int cb   = wave * 16 + m;       // B column inside a z row

  const float* zb = z + (size_t)b * N_DIM * E_DIM;

  // prologue: stage chunk 0 (rows 0..16)
  stage_rows(sbuf[0], zb, CH_ROWS, tid);
  wait_async0();
  __syncthreads();

  v8f accG = {0,0,0,0,0,0,0,0};
  v8f accC = {0,0,0,0,0,0,0,0};

  for (int ck = 0; ck < NCHUNK; ++ck) {
    const float* cur = sbuf[ck & 1];

    // issue next chunk's async copy BEFORE computing: DMA overlaps WMMA
    if (ck + 1 < NCHUNK) {
      const int nr = (ck + 1 == NCHUNK - 1) ? 16 : CH_ROWS; // tail: rows ..511
      stage_rows(sbuf[(ck + 1) & 1],
                 zb + (size_t)(ck + 1) * 16 * E_DIM, nr, tid);
    }

    const bool tail = (ck == NCHUNK - 1);
    #pragma unroll
    for (int k4 = 0; k4 < 16; k4 += 4) {
      const int s = k4 + kb;                  // k slot within chunk (0..14)
      const bool cut = tail && (s == 14);     // zero k==511 contributions
      v2f a, bg, bc;
      a.x  = cur[s * E_DIM + ca];
      a.y  = cut ? 0.f : cur[(s + 1) * E_DIM + ca];
      bg.x = cur[s * E_DIM + cb];
      bg.y = cut ? 0.f : cur[(s + 1) * E_DIM + cb];
      bc.x = cur[(s + 1) * E_DIM + cb];       // zf[k] = zb row k+1
      bc.y = cut ? 0.f : cur[(s + 2) * E_DIM + cb];
      accG = wmma_f32x4(a, bg, accG);
      accC = wmma_f32x4(a, bc, accC);
    }

    wait_async0();       // next chunk's data landed (overlapped with WMMAs)
    __syncthreads();
  }

  V8 uG, uC; uG.v = accG; uC.v = accC;
  const size_t base = (size_t)b * E_DIM * E_DIM;
  #pragma unroll
  for (int r = 0; r < 8; ++r) {
    const int row = ti * 16 + ((lane >> 4) << 3) + r;  // C/D layout: M=r+8*(lane/16)
    Gout[base + (size_t)row * E_DIM + cb] = uG.f[r];
    Cout[base + (size_t)row * E_DIM + cb] = uC.f[r];
  }
}

// ---------------------------------------------------------------------------
// Kernel 2: per-batch Gauss-Jordan on [G | C] in LDS -> K_local = G^-1 C,
// then K = K_local + K_global (written over G region), NaN -> flag.
// dyn LDS = (2*E*E + E) floats = 128.5 KB (WGP has 320 KB)
// ---------------------------------------------------------------------------
__global__ __launch_bounds__(256)
void solve_kernel(float* __restrict__ GK,          // in: G, out: K
                  const float* __restrict__ Cin,
                  const float* __restrict__ Kglob,
                  int* __restrict__ flag) {
  extern __shared__ float smem[];
  float* sG = smem;                    // E*E
  float* sC = smem + E_DIM * E_DIM;    // E*E
  float* ff = sC + E_DIM * E_DIM;      // E (elimination factors)

  const int b   = blockIdx.x;
  const int tid = threadIdx.x;
  const size_t base = (size_t)b * E_DIM * E_DIM;

  for (int i = tid; i < E_DIM * E_DIM; i += 256) {
    sG[i] = GK[base + i];
    sC[i] = Cin[base + i];
  }
  __syncthreads();

  for (int k = 0; k < E_DIM; ++k) {
    const float piv = sG[k * E_DIM + k];
    __syncthreads();                       // everyone reads pivot before writes
    const float inv = 1.0f / piv;
    if (tid < E_DIM) {
      sG[k * E_DIM + tid] *= inv;          // normalize row k (G half)
      ff[tid] = (tid == k) ? 0.0f : sG[tid * E_DIM + k];  // column-k factors
    } else {
      sC[k * E_DIM + (tid - 128)] *= inv;  // normalize row k (C half)
    }
    __syncthreads();
    if (tid < E_DIM) {
      for (int r = 0; r < E_DIM; ++r) {
        if (r == k) continue;
        const float f = ff[r];
        sG[r * E_DIM + tid] = fmaf(-f, sG[k * E_DIM + tid], sG[r * E_DIM + tid]);
      }
    } else {
      const int c2 = tid - 128;
      for (int r = 0; r < E_DIM; ++r) {
        if (r == k) continue;
        const float f = ff[r];
        sC[r * E_DIM + c2] = fmaf(-f, sC[k * E_DIM + c2], sC[r * E_DIM + c2]);
      }
    }
    __syncthreads();
  }

  int bad = 0;
  for (int i = tid; i < E_DIM * E_DIM; i += 256) {
    const float v = sC[i] + Kglob[i];       // K = K_local + K_global
    GK[base + i] = v;
    if (v != v) bad = 1;                    // NaN
  }
  if (bad) atomicOr(flag, 1);
}

// ---------------------------------------------------------------------------
// Kernel 3: rollout. Seed s_0..s_15 sequentially, build K^16 via 4 WMMA
// squarings in LDS, then advance 16 states per WMMA chunk-GEMM:
//   V_{c+1}(16xE) = V_c(16xE) * K16(ExE),  c = 1..37  (states up to s_607)
// dyn LDS = (2*E*E + 16*E + 256) floats ~= 137 KB
// ---------------------------------------------------------------------------
__device__ __forceinline__ void gemm128_square(const float* __restrict__ src,
                                               float* __restrict__ dst,
                                               int wave, int lane) {
  const int m   = lane & 15;
  const int kb  = (lane >> 4) << 1;
  const int col = wave * 16 + m;
  for (int mi = 0; mi < 8; ++mi) {
    v8f acc = {0,0,0,0,0,0,0,0};
    for (int t0 = 0; t0 < E_DIM; t0 += 4) {
      v2f a, bm;
      a.x  = src[(mi * 16 + m) * E_DIM + t0 + kb];
      a.y  = src[(mi * 16 + m) * E_DIM + t0 + kb + 1];
      bm.x = src[(t0 + kb)     * E_DIM + col];
      bm.y = src[(t0 + kb + 1) * E_DIM + col];
      acc = wmma_f32x4(a, bm, acc);
    }
    V8 u; u.v = acc;
    #pragma unroll
    for (int r = 0; r < 8; ++r)
      dst[(mi * 16 + ((lane >> 4) << 3) + r) * E_DIM + col] = u.f[r];
  }
}

__global__ __launch_bounds__(256)
void roll_kernel(const float* __restrict__ z,
                 const float* __restrict__ Kmat,
                 const int* __restrict__ flag,
                 float* __restrict__ rec,      // (B, 512, E)
                 float* __restrict__ fore) {   // (B, 96, E)
  extern __shared__ float smem[];
  float* sKa  = smem;                          // E*E : K, then K^4, then K^16
  float* sKb  = sKa + E_DIM * E_DIM;           // E*E : K^2, K^8
  float* V    = sKb + E_DIM * E_DIM;           // 16*E : current state chunk
  float* part = V + 16 * E_DIM;                // 256 : dot partials

  const int b    = blockIdx.x;
  const int tid  = threadIdx.x;
  const int wave = tid >> 5;
  const int lane = tid & 31;
  const size_t base = (size_t)b * E_DIM * E_DIM;
  const int fl = *flag;

  // load K (or identity fallback per reference's NaN-any check)
  for (int i = tid; i < E_DIM * E_DIM; i += 256)
    sKa[i] = fl ? ((i / E_DIM == i % E_DIM) ? 1.0f : 0.0f) : Kmat[base + i];
  if (tid < E_DIM) {
    const float x0 = z[(size_t)b * N_DIM * E_DIM + tid];
    V[tid] = x0;
    rec[(size_t)b * N_DIM * E_DIM + tid] = x0;   // x_rec[:,0,:] = z[:,0,:]
  }
  __syncthreads();

  // seed: s_1..s_15 sequentially (VALU dot products, 2 partials per output)
  for (int t = 1; t < 16; ++t) {
    const int j = tid & 127, h = tid >> 7;
    const float* xc = V + (t - 1) * E_DIM + h * 64;
    const float* kc = sKa + (size_t)h * 64 * E_DIM + j;
    float acc = 0.f;
    #pragma unroll 8
    for (int i = 0; i < 64; ++i) acc = fmaf(xc[i], kc[i * E_DIM], acc);
    part[tid] = acc;
    __syncthreads();
    if (tid < E_DIM) {
      const float y = part[tid] + part[tid + 128];
      V[t * E_DIM + tid] = y;
      rec[(size_t)b * N_DIM * E_DIM + (size_t)t * E_DIM + tid] = y;
    }
    __syncthreads();
  }

  // K^16 = (((K^2)^2)^2)^2 via WMMA squarings (ping-pong LDS)
  gemm128_square(sKa, sKb, wave, lane); __syncthreads();   // K^2
  gemm128_square(sKb, sKa, wave, lane); __syncthreads();   // K^4
  gemm128_square(sKa, sKb, wave, lane); __syncthreads();   // K^8
  gemm128_square(sKb, sKa, wave, lane); __syncthreads();   // K^16 in sKa

  // chunks: V_c = V_{c-1} * K^16 gives states s_{16c}..s_{16c+15}
  const int m  = lane & 15;
  const int kb = (lane >> 4) << 1;
  const int col = wave * 16 + m;
  for (int c = 1; c <= 37; ++c) {
    v8f acc = {0,0,0,0,0,0,0,0};
    for (int t0 = 0; t0 < E_DIM; t0 += 4) {
      v2f a, bm;
      a.x  = V[m * E_DIM + t0 + kb];
      a.y  = V[m * E_DIM + t0 + kb + 1];
      bm.x = sKa[(t0 + kb)     * E_DIM + col];
      bm.y = sKa[(t0 + kb + 1) * E_DIM + col];
      acc = wmma_f32x4(a, bm, acc);
    }
    __syncthreads();                 // all reads of V done before overwrite
    V8 u; u.v = acc;
    #pragma unroll
    for (int r = 0; r < 8; ++r) {
      const int row = ((lane >> 4) << 3) + r;      // state index within chunk
      const float val = u.f[r];
      V[row * E_DIM + col] = val;
      const int t = 16 * c + row;                  // global state index s_t
      if (t <= T_DIM)
        rec[(size_t)b * N_DIM * E_DIM + (size_t)t * E_DIM + col] = val;
      else if (t <= S_TOT)
        fore[(size_t)b * M_FORE * E_DIM + (size_t)(t - 512) * E_DIM + col] = val;
    }
    __syncthreads();
  }
}

// ---------------------------------------------------------------------------
extern "C" void kernel_launch(void* const* d_in, const int* in_sizes, int n_in,
                              void* d_out, int out_size, void* d_ws, size_t ws_size,
                              hipStream_t stream) {
  (void)in_sizes; (void)n_in; (void)out_size; (void)ws_size;
  const float* z  = (const float*)d_in[0];
  const float* Kg = (const float*)d_in[1];

  float* rec  = (float*)d_out;                               // B*512*E
  float* fore = rec + (size_t)B_DIM * N_DIM * E_DIM;         // B*96*E

  float* Gw   = (float*)d_ws;                                // B*E*E (G -> K)
  float* Cw   = Gw + (size_t)B_DIM * E_DIM * E_DIM;          // B*E*E
  int*   flag = (int*)(Cw + (size_t)B_DIM * E_DIM * E_DIM);  // 1 int

  hipLaunchKernelGGL(zero_flag_kernel, dim3(1), dim3(1), 0, stream, flag);

  hipLaunchKernelGGL(gram_kernel, dim3(B_DIM, E_DIM / 16), dim3(256), 0, stream,
                     z, Gw, Cw);

  const size_t solve_lds = (size_t)(2 * E_DIM * E_DIM + E_DIM) * sizeof(float);
  hipLaunchKernelGGL(solve_kernel, dim3(B_DIM), dim3(256), solve_lds, stream,
                     Gw, Cw, Kg, flag);

  const size_t roll_lds = (size_t)(2 * E_DIM * E_DIM + 16 * E_DIM + 256) * sizeof(float);
  hipLaunchKernelGGL(roll_kernel, dim3(B_DIM), dim3(256), roll_lds, stream,
                     z, Gw, flag, rec, fore);
}